// SGP_LAYER_23416161698459
// MI455X (gfx1250) — compile-verified
//
#include <hip/hip_runtime.h>
#include <cstddef>

// ---------------------------------------------------------------------------
// Problem constants (from the reference)
// ---------------------------------------------------------------------------
constexpr int kB  = 2;
constexpr int kL  = 256;
constexpr int kHD = 512;
constexpr int kH  = 8;
constexpr int kM  = 64;
constexpr int kS  = 2;
constexpr int kDQ = 64;
constexpr float kJit = 1e-3f;

typedef __attribute__((ext_vector_type(2))) float v2f;
typedef __attribute__((ext_vector_type(8))) float v8f;

// CDNA5 fp32 matrix core: D(16x16) = A(16x4) * B(4x16) + C
__device__ __forceinline__ v8f wmma4(v2f a, v2f b, v8f c) {
  return __builtin_amdgcn_wmma_f32_16x16x4_f32(false, a, false, b, (short)0, c,
                                               false, false);
}

// Optional CDNA5 async global->LDS path (guarded; falls back to plain copies)
#if defined(__has_builtin)
#if __has_builtin(__builtin_amdgcn_global_load_async_to_lds_b32) && \
    __has_builtin(__builtin_amdgcn_s_wait_asynccnt)
#define USE_ASYNC_LDS 1
#endif
#endif

#ifdef USE_ASYNC_LDS
__device__ __forceinline__ void async_g2l_b32(const float* g, float* l) {
  // prototype (from compiler diagnostic): (int AS1*, int AS3*, imm off, imm cpol)
  __builtin_amdgcn_global_load_async_to_lds_b32(
      (__attribute__((address_space(1))) int*)(g),
      (__attribute__((address_space(3))) int*)(l), 0, 0);
}
#endif

// Fragment addressing (one wave = one 16x16 C tile):
//   lane = l16 + 16*hi;  A frag: a[j] = A[m0+l16][k0+2*hi+j]
//                        B frag: b[j] = B[k0+2*hi+j][n0+l16]
//   C frag: c[r] -> C[m0 + r + 8*hi][n0 + l16]

// ---------------------------------------------------------------------------
// 1) q / v_gamma projections:  q[b,h,l,d] = sum_c x[b,l,c] * W[h*64+d, c]
// grid: x = 64 tiles (16 M-tiles x 4 N-tiles), y = B*H, z = {Wqk->q, Wv->vg}
// ---------------------------------------------------------------------------
__global__ void k_proj(const float* __restrict__ x,
                       const float* __restrict__ Wqk,
                       const float* __restrict__ Wv,
                       float* __restrict__ q, float* __restrict__ vg) {
  const int mt = blockIdx.x >> 2, nt = blockIdx.x & 3;
  const int batch = blockIdx.y;
  const int b = batch / kH, h = batch % kH;
  const float* W = (blockIdx.z == 0) ? Wqk : Wv;
  float* out = (blockIdx.z == 0) ? q : vg;
  const int lane = threadIdx.x & 31, l16 = lane & 15, hi = lane >> 4;
  const int m0 = mt * 16, n0 = nt * 16;
  const float* A  = x + (size_t)b * kL * kHD;
  const float* Bp = W + (size_t)(h * kDQ + n0) * kHD;  // row (h*64+n0+l16)
  v8f c = {};
  for (int k0 = 0; k0 < kHD; k0 += 4) {
    const int kk = k0 + 2 * hi;
    __builtin_prefetch(&A[(m0 + l16) * kHD + k0 + 64], 0, 3);
    v2f a, bf;
    a.x  = A[(m0 + l16) * kHD + kk];
    a.y  = A[(m0 + l16) * kHD + kk + 1];
    bf.x = Bp[l16 * kHD + kk];
    bf.y = Bp[l16 * kHD + kk + 1];
    c = wmma4(a, bf, c);
  }
  float* O = out + ((size_t)batch * kL) * kDQ;
#pragma unroll
  for (int r = 0; r < 8; ++r)
    O[(m0 + r + 8 * hi) * kDQ + n0 + l16] = c[r];
}

// ---------------------------------------------------------------------------
// 2) k_beta[h,m,d] = sum_c W_qk[h*64+d, c] * cur_k[h,0,m,c]
// grid: x = 16 tiles (4x4), y = H
// ---------------------------------------------------------------------------
__global__ void k_kbeta(const float* __restrict__ cur_k,
                        const float* __restrict__ Wqk,
                        float* __restrict__ kbeta) {
  const int mt = blockIdx.x >> 2, nt = blockIdx.x & 3;
  const int h = blockIdx.y;
  const int lane = threadIdx.x & 31, l16 = lane & 15, hi = lane >> 4;
  const int m0 = mt * 16, n0 = nt * 16;
  const float* A  = cur_k + (size_t)h * kM * kHD;
  const float* Bp = Wqk + (size_t)(h * kDQ + n0) * kHD;
  v8f c = {};
  for (int k0 = 0; k0 < kHD; k0 += 4) {
    const int kk = k0 + 2 * hi;
    v2f a, bf;
    a.x  = A[(m0 + l16) * kHD + kk];
    a.y  = A[(m0 + l16) * kHD + kk + 1];
    bf.x = Bp[l16 * kHD + kk];
    bf.y = Bp[l16 * kHD + kk + 1];
    c = wmma4(a, bf, c);
  }
#pragma unroll
  for (int r = 0; r < 8; ++r)
    kbeta[((size_t)h * kM + m0 + r + 8 * hi) * kDQ + n0 + l16] = c[r];
}

// ---------------------------------------------------------------------------
// 3) length-scale scaling + row sum-of-squares for q and k_beta
// ---------------------------------------------------------------------------
__global__ void __launch_bounds__(256) k_scale(
    const float* __restrict__ q, const float* __restrict__ kbeta,
    const float* __restrict__ log_ls, float* __restrict__ qs,
    float* __restrict__ qsq, float* __restrict__ ksb,
    float* __restrict__ ksq) {
  const int idx = blockIdx.x * 256 + threadIdx.x;
  const int nq = kB * kH * kL;  // 4096
  const int nk = kH * kM;       // 512
  if (idx < nq) {
    const int h = (idx / kL) % kH;
    float s = 0.f;
    for (int d = 0; d < kDQ; ++d) {
      const float inv = __expf(-log_ls[h * kDQ + d]);
      const float val = q[(size_t)idx * kDQ + d] * inv;
      qs[(size_t)idx * kDQ + d] = val;
      s += val * val;
    }
    qsq[idx] = s;
  } else if (idx < nq + nk) {
    const int j = idx - nq;
    const int h = j / kM;
    float s = 0.f;
    for (int d = 0; d < kDQ; ++d) {
      const float inv = __expf(-log_ls[h * kDQ + d]);
      const float val = kbeta[(size_t)j * kDQ + d] * inv;
      ksb[(size_t)j * kDQ + d] = val;
      s += val * val;
    }
    ksq[j] = s;
  }
}

// ---------------------------------------------------------------------------
// 4) RBF Gram matrices: out = sf * exp(-0.5*max(|a|^2+|b|^2-2 a.b, 0))
//    G = A (Mx64) @ B^T (64xN) via WMMA, exp epilogue.
// ---------------------------------------------------------------------------
__global__ void k_rbf(const float* __restrict__ A, const float* __restrict__ Asq,
                      const float* __restrict__ Bm, const float* __restrict__ Bsq,
                      const float* __restrict__ log_sf, float* __restrict__ out,
                      int Ncols, int aBS, int asqBS, int bBS, int bsqBS, int oBS,
                      int aModH, int bModH) {
  const int tilesN = Ncols >> 4;
  const int mt = blockIdx.x / tilesN;
  const int nt = blockIdx.x % tilesN;
  const int batch = blockIdx.y;
  const int h = batch % kH;
  const int ai = aModH ? h : batch;
  const int bi = bModH ? h : batch;
  const float* Ap = A + (size_t)ai * aBS;
  const float* Bp = Bm + (size_t)bi * bBS;
  const float* AsqP = Asq + (size_t)ai * asqBS;
  const float* BsqP = Bsq + (size_t)bi * bsqBS;
  const int lane = threadIdx.x & 31, l16 = lane & 15, hi = lane >> 4;
  const int m0 = mt * 16, n0 = nt * 16;
  v8f c = {};
#pragma unroll
  for (int k0 = 0; k0 < kDQ; k0 += 4) {
    const int kk = k0 + 2 * hi;
    v2f a, bf;
    a.x  = Ap[(m0 + l16) * kDQ + kk];
    a.y  = Ap[(m0 + l16) * kDQ + kk + 1];
    bf.x = Bp[(n0 + l16) * kDQ + kk];   // B[k][n] = Bp[n*64+k]
    bf.y = Bp[(n0 + l16) * kDQ + kk + 1];
    c = wmma4(a, bf, c);
  }
  const float sf = __expf(log_sf[h]);
  float* Op = out + (size_t)batch * oBS;
#pragma unroll
  for (int r = 0; r < 8; ++r) {
    const int mr = m0 + r + 8 * hi, nc = n0 + l16;
    const float sq = AsqP[mr] + BsqP[nc] - 2.0f * c[r];
    Op[mr * Ncols + nc] = sf * __expf(-0.5f * fmaxf(sq, 0.0f));
  }
}

// ---------------------------------------------------------------------------
// 5) out = Kqq - X @ X^T  with X (256x64).  Used for Amat (X=Kqb) and for
//    base = Kqq - v1^T v1 (X=v1T, since v2 == v1 by kernel symmetry).
// grid: x = 256 tiles, y = B*H
// ---------------------------------------------------------------------------
__global__ void k_amat(const float* __restrict__ Kqq, const float* __restrict__ X,
                       float* __restrict__ out) {
  const int mt = blockIdx.x >> 4, nt = blockIdx.x & 15;
  const int batch = blockIdx.y;
  const float* Xb = X + (size_t)batch * kL * kM;
  const float* Kb = Kqq + (size_t)batch * kL * kL;
  const int lane = threadIdx.x & 31, l16 = lane & 15, hi = lane >> 4;
  const int m0 = mt * 16, n0 = nt * 16;
  v8f c = {};
#pragma unroll
  for (int k0 = 0; k0 < kM; k0 += 4) {
    const int kk = k0 + 2 * hi;
    v2f a, bf;
    a.x  = Xb[(m0 + l16) * kM + kk];
    a.y  = Xb[(m0 + l16) * kM + kk + 1];
    bf.x = Xb[(n0 + l16) * kM + kk];
    bf.y = Xb[(n0 + l16) * kM + kk + 1];
    c = wmma4(a, bf, c);
  }
  float* Ob = out + (size_t)batch * kL * kL;
#pragma unroll
  for (int r = 0; r < 8; ++r) {
    const int mr = m0 + r + 8 * hi, nc = n0 + l16;
    Ob[mr * kL + nc] = Kb[mr * kL + nc] - c[r];
  }
}

// ---------------------------------------------------------------------------
// 6) mean = Amat @ v_gamma + Kqb @ v_beta      (256x64, K=256 then K=64)
// grid: x = 64 tiles (16x4), y = B*H
// ---------------------------------------------------------------------------
__global__ void k_mean(const float* __restrict__ Amat, const float* __restrict__ vg,
                       const float* __restrict__ Kqb, const float* __restrict__ vp,
                       float* __restrict__ mean) {
  const int mt = blockIdx.x >> 2, nt = blockIdx.x & 3;
  const int batch = blockIdx.y;
  const int h = batch % kH;
  const int lane = threadIdx.x & 31, l16 = lane & 15, hi = lane >> 4;
  const int m0 = mt * 16, n0 = nt * 16;
  const float* Ab = Amat + (size_t)batch * kL * kL;
  const float* Vb = vg + (size_t)batch * kL * kDQ;
  const float* Qb = Kqb + (size_t)batch * kL * kM;
  const float* Pb = vp + (size_t)h * kM * kDQ;  // v_beta[h,m,v] = v_param[h,0,m,v]
  v8f c = {};
  for (int k0 = 0; k0 < kL; k0 += 4) {
    const int kk = k0 + 2 * hi;
    v2f a, bf;
    a.x  = Ab[(m0 + l16) * kL + kk];
    a.y  = Ab[(m0 + l16) * kL + kk + 1];
    bf.x = Vb[kk * kDQ + n0 + l16];
    bf.y = Vb[(kk + 1) * kDQ + n0 + l16];
    c = wmma4(a, bf, c);
  }
#pragma unroll
  for (int k0 = 0; k0 < kM; k0 += 4) {
    const int kk = k0 + 2 * hi;
    v2f a, bf;
    a.x  = Qb[(m0 + l16) * kM + kk];
    a.y  = Qb[(m0 + l16) * kM + kk + 1];
    bf.x = Pb[kk * kDQ + n0 + l16];
    bf.y = Pb[(kk + 1) * kDQ + n0 + l16];
    c = wmma4(a, bf, c);
  }
  float* Ob = mean + (size_t)batch * kL * kDQ;
#pragma unroll
  for (int r = 0; r < 8; ++r)
    Ob[(m0 + r + 8 * hi) * kDQ + n0 + l16] = c[r];
}

// ---------------------------------------------------------------------------
// 7) 64x64 Cholesky of Kbb + jitter, one block (64 threads) per head
// ---------------------------------------------------------------------------
__global__ void __launch_bounds__(64) k_cholbb(const float* __restrict__ Kbb,
                                               float* __restrict__ cholbb) {
  __shared__ float A[kM * kM];
  const int h = blockIdx.x, t = threadIdx.x;
  for (int j = 0; j < kM; ++j)
    A[t * kM + j] = Kbb[(size_t)h * kM * kM + t * kM + j] + ((t == j) ? kJit : 0.f);
  __syncthreads();
  for (int k = 0; k < kM; ++k) {
    if (t == k) A[k * kM + k] = sqrtf(A[k * kM + k]);
    __syncthreads();
    const float dk = A[k * kM + k];
    if (t > k) A[t * kM + k] /= dk;
    __syncthreads();
    if (t > k) {
      const float lik = A[t * kM + k];
      for (int j = k + 1; j <= t; ++j) A[t * kM + j] -= lik * A[j * kM + k];
    }
    __syncthreads();
  }
  for (int j = 0; j < kM; ++j)
    cholbb[(size_t)h * kM * kM + t * kM + j] = (j <= t) ? A[t * kM + j] : 0.f;
}

// ---------------------------------------------------------------------------
// 8) v1 (== v2): forward substitution  chol_bb * y = Kqb[l,:]^T, one column
//    per thread, stored row-major as v1T[b,h,l,m].
// grid: x = B*H, 256 threads
// ---------------------------------------------------------------------------
__global__ void __launch_bounds__(256) k_v1(const float* __restrict__ cholbb,
                                            const float* __restrict__ Kqb,
                                            float* __restrict__ v1T) {
  const int bh = blockIdx.x;
  const int h = bh % kH;
  const int l = threadIdx.x;
  __shared__ float Lc[kM * kM];
  for (int i = threadIdx.x; i < kM * kM; i += 256)
    Lc[i] = cholbb[(size_t)h * kM * kM + i];
  __syncthreads();
  const float* R = Kqb + (size_t)bh * kL * kM + (size_t)l * kM;
  float y[kM];
#pragma unroll
  for (int i = 0; i < kM; ++i) {
    float s = R[i];
#pragma unroll
    for (int j = 0; j < i; ++j) s -= Lc[i * kM + j] * y[j];
    y[i] = s / Lc[i * kM + i];
  }
  float* o = v1T + (size_t)bh * kL * kM + (size_t)l * kM;
#pragma unroll
  for (int m = 0; m < kM; ++m) o[m] = y[m];
}

// ---------------------------------------------------------------------------
// 9) KL quadratic term: 0.5/B * sum_b sum_h sum_{l,k} base * (vg vg^T)
// ---------------------------------------------------------------------------
__global__ void __launch_bounds__(256) k_kltemp(const float* __restrict__ base,
                                                const float* __restrict__ vg,
                                                float* __restrict__ kl) {
  const int bh = blockIdx.x;
  const int l = threadIdx.x;
  const float* Bp = base + (size_t)bh * kL * kL;
  const float* V  = vg + (size_t)bh * kL * kDQ;
  float acc = 0.f;
  for (int k = 0; k < kL; ++k) {
    float d = 0.f;
    for (int v = 0; v < kDQ; ++v) d += V[l * kDQ + v] * V[k * kDQ + v];
    acc += Bp[l * kL + k] * d;
  }
  __shared__ float red[256];
  red[l] = acc;
  __syncthreads();
  for (int st = 128; st > 0; st >>= 1) {
    if (l < st) red[l] += red[l + st];
    __syncthreads();
  }
  if (l == 0) atomicAdd(kl, (0.5f / (float)kB) * red[0]);
}

// ---------------------------------------------------------------------------
// 10) KL small terms + materialize s_local (H,DQ,M,M)
// ---------------------------------------------------------------------------
__global__ void __launch_bounds__(256) k_klsmall(
    const float* __restrict__ ltri, const float* __restrict__ lssd,
    const float* __restrict__ Kbb, const float* __restrict__ vp,
    float* __restrict__ sloc, float* __restrict__ kl) {
  const int t = threadIdx.x;
  float acc = 0.f;
  const int total = kH * kDQ * kM * kM;
  for (int i = t; i < total; i += 256) {
    const int n = i & (kM - 1);
    const int m = (i >> 6) & (kM - 1);
    const int hv = i >> 12;
    float val;
    if (n < m)       val = ltri[i];
    else if (n == m) val = __expf(lssd[hv * kM + m]);
    else             val = 0.f;
    sloc[i] = val;
    acc += 0.5f * val * val;
  }
  for (int i = t; i < kH * kDQ * kM; i += 256) acc -= lssd[i];
  for (int i = t; i < kH * kM * kM; i += 256) {
    const int h = i >> 12, m = (i >> 6) & 63, n = i & 63;
    float d = 0.f;
    for (int v = 0; v < kDQ; ++v)
      d += vp[((size_t)h * kM + m) * kDQ + v] * vp[((size_t)h * kM + n) * kDQ + v];
    acc += 0.5f * Kbb[i] * d;
  }
  __shared__ float red[256];
  red[t] = acc;
  __syncthreads();
  for (int st = 128; st > 0; st >>= 1) {
    if (t < st) red[t] += red[t + st];
    __syncthreads();
  }
  if (t == 0) atomicAdd(kl, red[0]);
}

__global__ void k_init(float* kl) { *kl = -0.5f * kM * kDQ * kH; }

// ---------------------------------------------------------------------------
// 11) FUSED sampler: one workgroup per (b,h,v) slice.
//     LDS: cov (256x256 fp32, 256KB) + v3 (256x64, 64KB) = 320KB (full WGP LDS)
//     a) stage s_local via ASYNC global->LDS, v3 = v1^T @ s_local_v
//     b) cov = base + v3 v3^T + jit*I     (WMMA from LDS)
//     c) async-stage eps rows overlapped with in-place Cholesky in LDS
//     d) noise = tril(cov) @ eps; samples = mean + noise
//     The 256x256 covariance/cholesky never touches HBM (saves ~540MB traffic).
// ---------------------------------------------------------------------------
__global__ void __launch_bounds__(256) k_sample(
    const float* __restrict__ base, const float* __restrict__ v1T,
    const float* __restrict__ sloc, const float* __restrict__ mean,
    const float* __restrict__ eps, float* __restrict__ smp) {
  extern __shared__ float lds[];
  float* cov = lds;              // 65536 floats
  float* v3  = lds + kL * kL;    // 16384 floats
  const int bhv = blockIdx.x;
  const int v = bhv % kDQ;
  const int bh = bhv / kDQ;
  const int h = bh % kH;
  const int b = bh / kH;
  const int tid = threadIdx.x;

  const float* bb = base + (size_t)bh * kL * kL;
  const float* v1 = v1T + (size_t)bh * kL * kM;
  const float* sl = sloc + ((size_t)(h * kDQ + v)) * kM * kM;

  // (a0) stage s_local (64x64 = 16KB) into LDS; cov region is free right now
  float* slsh = cov;
#ifdef USE_ASYNC_LDS
  for (int i = tid; i < kM * kM; i += 256) async_g2l_b32(sl + i, slsh + i);
  __builtin_amdgcn_s_wait_asynccnt(0);
#else
  for (int i = tid; i < kM * kM; i += 256) slsh[i] = sl[i];
#endif
  __syncthreads();

  // (a) v3[l][n] = sum_m v1T[l,m] * s_local[h,v,m,n]; thread = row l
  {
    float vrow[kM];
#pragma unroll
    for (int m = 0; m < kM; ++m) vrow[m] = v1[(size_t)tid * kM + m];
    for (int n = 0; n < kM; ++n) {
      float s = 0.f;
#pragma unroll 8
      for (int m = 0; m < kM; ++m) s += vrow[m] * slsh[m * kM + n];
      v3[tid * kM + n] = s;
    }
  }
  __syncthreads();

  // (b) cov = base + v3 @ v3^T + jitter*I  via WMMA (fragments from LDS)
  const int lane = tid & 31, wave = tid >> 5;
  const int l16 = lane & 15, hi = lane >> 4;
  for (int t2 = wave; t2 < 256; t2 += 8) {
    const int mt = t2 >> 4, nt = t2 & 15;
    const int m0 = mt * 16, n0 = nt * 16;
    v8f c = {};
#pragma unroll
    for (int k0 = 0; k0 < kM; k0 += 4) {
      const int kk = k0 + 2 * hi;
      v2f a, bf;
      a.x  = v3[(m0 + l16) * kM + kk];
      a.y  = v3[(m0 + l16) * kM + kk + 1];
      bf.x = v3[(n0 + l16) * kM + kk];
      bf.y = v3[(n0 + l16) * kM + kk + 1];
      c = wmma4(a, bf, c);
    }
#pragma unroll
    for (int r = 0; r < 8; ++r) {
      const int mr = m0 + r + 8 * hi, nc = n0 + l16;
      float val = bb[(size_t)mr * kL + nc] + c[r];
      if (mr == nc) val += kJit;
      cov[mr * kL + nc] = val;
    }
  }
  __syncthreads();

  // (c0) kick off async staging of the two eps rows into the retired v3
  //      region; the copy overlaps the entire Cholesky below.
  float* epsh = v3;  // [s*kL + k], 512 floats
  const float* ep0 = eps + ((((size_t)b * kH + h) * kS + 0) * kDQ + v) * kL;
  const float* ep1 = ep0 + (size_t)kDQ * kL;
#ifdef USE_ASYNC_LDS
  async_g2l_b32(ep0 + tid, epsh + tid);
  async_g2l_b32(ep1 + tid, epsh + kL + tid);
#else
  epsh[tid] = ep0[tid];
  epsh[kL + tid] = ep1[tid];
#endif

  // (c) in-place Cholesky (lower), thread tid owns row tid
  for (int k = 0; k < kL; ++k) {
    if (tid == k) cov[k * kL + k] = sqrtf(cov[k * kL + k]);
    __syncthreads();
    const float dk = cov[k * kL + k];
    if (tid > k) cov[tid * kL + k] /= dk;
    __syncthreads();
    if (tid > k) {
      const float lik = cov[tid * kL + k];
      for (int j = k + 1; j <= tid; ++j)
        cov[tid * kL + j] -= lik * cov[j * kL + k];
    }
    __syncthreads();
  }

#ifdef USE_ASYNC_LDS
  __builtin_amdgcn_s_wait_asynccnt(0);
#endif
  __syncthreads();

  // (d) noise + samples: samples[b,h,s,l,v] = mean[b,h,l,v] + tril(cov)@eps
  const float* mb = mean + (size_t)bh * kL * kDQ;
  for (int s = 0; s < kS; ++s) {
    float acc = 0.f;
    for (int k = 0; k <= tid; ++k) acc += cov[tid * kL + k] * epsh[s * kL + k];
    smp[((((size_t)bh) * kS + s) * kL + tid) * kDQ + v] =
        mb[(size_t)tid * kDQ + v] + acc;
  }
}

// ---------------------------------------------------------------------------
// 12) output projection: out[b,s,l,:] = samples(b,s,l,·) @ Wo_w^T + Wo_b
// grid: x = 512 tiles (16 x 32), y = B*S
// ---------------------------------------------------------------------------
__global__ void k_outproj(const float* __restrict__ smp,
                          const float* __restrict__ Wo_w,
                          const float* __restrict__ Wo_b,
                          float* __restrict__ out) {
  const int mt = blockIdx.x >> 5, nt = blockIdx.x & 31;
  const int bs = blockIdx.y;
  const int b = bs / kS, s = bs % kS;
  const int lane = threadIdx.x & 31, l16 = lane & 15, hi = lane >> 4;
  const int m0 = mt * 16, n0 = nt * 16;
  const int mrow = m0 + l16;
  v8f c = {};
  for (int k0 = 0; k0 < kHD; k0 += 4) {
    const int kk = k0 + 2 * hi;
    const int c0 = kk, c1 = kk + 1;
    v2f a, bf;
    // channel ck = h*64 + vv lives at smp[(((b*H + h)*S + s)*L + l)*64 + vv]
    a.x = smp[((((size_t)b * kH + (c0 >> 6)) * kS + s) * kL + mrow) * kDQ + (c0 & 63)];
    a.y = smp[((((size_t)b * kH + (c1 >> 6)) * kS + s) * kL + mrow) * kDQ + (c1 & 63)];
    bf.x = Wo_w[(size_t)(n0 + l16) * kHD + c0];
    bf.y = Wo_w[(size_t)(n0 + l16) * kHD + c1];
    c = wmma4(a, bf, c);
  }
  float* Ob = out + ((size_t)bs * kL) * kHD;
  const float bias = Wo_b[n0 + l16];
#pragma unroll
  for (int r = 0; r < 8; ++r)
    Ob[(m0 + r + 8 * hi) * kHD + n0 + l16] = c[r] + bias;
}

// ---------------------------------------------------------------------------
// Host launcher
// ---------------------------------------------------------------------------
extern "C" void kernel_launch(void* const* d_in, const int* in_sizes, int n_in,
                              void* d_out, int out_size, void* d_ws,
                              size_t ws_size, hipStream_t stream) {
  (void)in_sizes; (void)n_in; (void)out_size; (void)ws_size;
  const float* x       = (const float*)d_in[0];
  const float* cur_k   = (const float*)d_in[1];
  const float* eps     = (const float*)d_in[2];
  const float* W_qk    = (const float*)d_in[3];
  const float* W_v     = (const float*)d_in[4];
  const float* log_ls  = (const float*)d_in[5];
  const float* log_sf  = (const float*)d_in[6];
  const float* v_param = (const float*)d_in[7];
  const float* ltri    = (const float*)d_in[8];
  const float* log_ssd = (const float*)d_in[9];
  const float* Wo_w    = (const float*)d_in[10];
  const float* Wo_b    = (const float*)d_in[11];

  float* out = (float*)d_out;
  float* kl  = out + (size_t)kB * kS * kL * kHD;  // scalar at the end

  // workspace layout (floats)
  float* ws     = (float*)d_ws;
  float* q      = ws;                 // 262144
  float* vg     = q      + 262144;    // 262144
  float* qs     = vg     + 262144;    // 262144
  float* qsq    = qs     + 262144;    // 4096
  float* kbeta  = qsq    + 4096;      // 32768
  float* ksb    = kbeta  + 32768;     // 32768
  float* ksq    = ksb    + 32768;     // 512
  float* Kqq    = ksq    + 512;       // 2097152
  float* Kqb    = Kqq    + 2097152;   // 524288
  float* Kbb    = Kqb    + 524288;    // 32768
  float* AB     = Kbb    + 32768;     // 2097152 (Amat, then base)
  float* meanb  = AB     + 2097152;   // 262144
  float* cholbb = meanb  + 262144;    // 32768
  float* v1T    = cholbb + 32768;     // 524288
  float* sloc   = v1T    + 524288;    // 2097152
  float* smp    = sloc   + 2097152;   // 524288
  // total ~9.05M floats (~34.5 MB)

  const int kSampleLds = (kL * kL + kL * kM) * (int)sizeof(float);  // 320 KB
  (void)hipFuncSetAttribute((const void*)k_sample,
                            hipFuncAttributeMaxDynamicSharedMemorySize,
                            kSampleLds);

  k_init<<<dim3(1), 1, 0, stream>>>(kl);
  k_proj<<<dim3(64, kB * kH, 2), 32, 0, stream>>>(x, W_qk, W_v, q, vg);
  k_kbeta<<<dim3(16, kH), 32, 0, stream>>>(cur_k, W_qk, kbeta);
  k_scale<<<dim3(18), 256, 0, stream>>>(q, kbeta, log_ls, qs, qsq, ksb, ksq);

  // K_qq (256x256, per b,h)
  k_rbf<<<dim3(256, kB * kH), 32, 0, stream>>>(qs, qsq, qs, qsq, log_sf, Kqq,
      kL, kL * kDQ, kL, kL * kDQ, kL, kL * kL, 0, 0);
  // K_qb (256x64, per b,h; beta side indexed by head)
  k_rbf<<<dim3(64, kB * kH), 32, 0, stream>>>(qs, qsq, ksb, ksq, log_sf, Kqb,
      kM, kL * kDQ, kL, kM * kDQ, kM, kL * kM, 0, 1);
  // K_bb (64x64, per head)
  k_rbf<<<dim3(16, kH), 32, 0, stream>>>(ksb, ksq, ksb, ksq, log_sf, Kbb,
      kM, kM * kDQ, kM, kM * kDQ, kM, kM * kM, 0, 0);

  k_amat<<<dim3(256, kB * kH), 32, 0, stream>>>(Kqq, Kqb, AB);      // Amat
  k_mean<<<dim3(64, kB * kH), 32, 0, stream>>>(AB, vg, Kqb, v_param, meanb);
  k_cholbb<<<dim3(kH), 64, 0, stream>>>(Kbb, cholbb);
  k_v1<<<dim3(kB * kH), 256, 0, stream>>>(cholbb, Kqb, v1T);
  k_amat<<<dim3(256, kB * kH), 32, 0, stream>>>(Kqq, v1T, AB);      // base
  k_kltemp<<<dim3(kB * kH), 256, 0, stream>>>(AB, vg, kl);
  k_klsmall<<<dim3(1), 256, 0, stream>>>(ltri, log_ssd, Kbb, v_param, sloc, kl);
  k_sample<<<dim3(kB * kH * kDQ), 256, kSampleLds, stream>>>(AB, v1T, sloc,
                                                             meanb, eps, smp);
  k_outproj<<<dim3(512, kB * kS), 32, 0, stream>>>(smp, Wo_w, Wo_b, out);
}